// GAT_120259084717
// MI455X (gfx1250) — compile-verified
//
#include <hip/hip_runtime.h>
#include <hip/hip_bf16.h>

typedef __attribute__((ext_vector_type(2))) float v2f;
typedef __attribute__((ext_vector_type(8))) float v8f;

#define INCH  128   // layer-1 in channels
#define HC    256   // H*C hidden
#define NHD   8     // heads layer 1
#define CPH   32    // channels per head
#define OUTC  40    // layer-2 out channels
#define SLOPE 0.2f

__device__ __forceinline__ float leaky(float x) { return x > 0.f ? x : SLOPE * x; }

// ---------------- GEMM1: H1[N,256] = X[N,128] @ W1[128,256] (fp32 WMMA) -------------
// block = 256 thr = 8 waves; each wave: one 16-row tile x two 16-col tiles.
__global__ void gemm_x_w1(const float* __restrict__ X, const float* __restrict__ W,
                          float* __restrict__ H1, int nrows) {
    const int lane = threadIdx.x & 31;
    const int wave = threadIdx.x >> 5;
    const int m    = lane & 15;     // A row / B,C col within tile
    const int half = lane >> 4;
    const int rowBase = blockIdx.x << 4;
    if (rowBase >= nrows) return;
    const float* xrow = X + (size_t)(rowBase + m) * INCH;
    for (int t = 0; t < 2; ++t) {
        const int colBase = ((wave << 1) + t) << 4;
        v8f acc = {0.f, 0.f, 0.f, 0.f, 0.f, 0.f, 0.f, 0.f};
        for (int k = 0; k < INCH; k += 4) {
            const int kk = k + (half << 1);          // K index for VGPR v: kk + v
            v2f a, b;
            a.x = xrow[kk];
            a.y = xrow[kk + 1];
            b.x = W[(size_t)kk * HC + colBase + m];
            b.y = W[(size_t)(kk + 1) * HC + colBase + m];
            acc = __builtin_amdgcn_wmma_f32_16x16x4_f32(false, a, false, b,
                                                        (short)0, acc, false, false);
        }
#pragma unroll
        for (int v = 0; v < 8; ++v)
            H1[(size_t)(rowBase + v + (half << 3)) * HC + colBase + m] = acc[v];
    }
}

// ---------------- GEMM2: H2[N,40] = OUT1[N,256] @ W2[256,40] (fp32 WMMA) ------------
// block = 96 thr = 3 waves; wave w -> col tile w (cols padded 40->48, guarded).
__global__ void gemm_h_w2(const float* __restrict__ Hin, const float* __restrict__ W,
                          float* __restrict__ H2, int nrows) {
    const int lane = threadIdx.x & 31;
    const int wave = threadIdx.x >> 5;
    const int m    = lane & 15;
    const int half = lane >> 4;
    const int rowBase = blockIdx.x << 4;
    if (rowBase >= nrows) return;
    const int colBase = wave << 4;
    const int col = colBase + m;
    const float* hrow = Hin + (size_t)(rowBase + m) * HC;
    v8f acc = {0.f, 0.f, 0.f, 0.f, 0.f, 0.f, 0.f, 0.f};
    for (int k = 0; k < HC; k += 4) {
        const int kk = k + (half << 1);
        v2f a, b;
        a.x = hrow[kk];
        a.y = hrow[kk + 1];
        b.x = (col < OUTC) ? W[kk * OUTC + col] : 0.f;
        b.y = (col < OUTC) ? W[(kk + 1) * OUTC + col] : 0.f;
        acc = __builtin_amdgcn_wmma_f32_16x16x4_f32(false, a, false, b,
                                                    (short)0, acc, false, false);
    }
    if (col < OUTC) {
#pragma unroll
        for (int v = 0; v < 8; ++v)
            H2[(size_t)(rowBase + v + (half << 3)) * OUTC + col] = acc[v];
    }
}

// ---------------- per-node attention dots, layer 1 (LDS-staged att vectors) ---------
__global__ void attn_dots_l1(const float* __restrict__ H1, const float* __restrict__ Asrc,
                             const float* __restrict__ Adst, float* __restrict__ aS,
                             float* __restrict__ aD, int n) {
    __shared__ float sAs[HC], sAd[HC];
    for (int i = threadIdx.x; i < HC; i += blockDim.x) { sAs[i] = Asrc[i]; sAd[i] = Adst[i]; }
    __syncthreads();
    const int idx = blockIdx.x * blockDim.x + threadIdx.x;
    if (idx >= n * NHD) return;
    const int node = idx >> 3, h = idx & 7;
    const float* row = H1 + (size_t)node * HC + h * CPH;
    float s = 0.f, d = 0.f;
#pragma unroll
    for (int c = 0; c < CPH; ++c) {
        const float v = row[c];
        s += v * sAs[h * CPH + c];
        d += v * sAd[h * CPH + c];
    }
    aS[idx] = s; aD[idx] = d;
}

__global__ void attn_dots_l2(const float* __restrict__ H2, const float* __restrict__ Asrc,
                             const float* __restrict__ Adst, float* __restrict__ aS,
                             float* __restrict__ aD, int n) {
    const int node = blockIdx.x * blockDim.x + threadIdx.x;
    if (node >= n) return;
    const float* row = H2 + (size_t)node * OUTC;
    float s = 0.f, d = 0.f;
#pragma unroll
    for (int c = 0; c < OUTC; ++c) { const float v = row[c]; s += v * Asrc[c]; d += v * Adst[c]; }
    aS[node] = s; aD[node] = d;
}

// ======================= CSR-by-destination build (built once) ======================
__global__ void count_edges(const int* __restrict__ dstA, int* __restrict__ cnt, int E, int n) {
    const int e = blockIdx.x * blockDim.x + threadIdx.x;
    if (e >= E + n) return;
    const int d = (e < E) ? dstA[e] : (e - E);     // self-loops appended
    atomicAdd(&cnt[d], 1);
}

// single-block exclusive scan (1024 thr, Hillis-Steele in LDS, sequential chunks)
__global__ void scan_rowptr(const int* __restrict__ cnt, int* __restrict__ rowptr,
                            int* __restrict__ cursor, int n) {
    __shared__ int lds[1024];
    __shared__ int carryS;
    int carry = 0;
    for (int base = 0; base < n; base += 1024) {
        const int i = base + (int)threadIdx.x;
        const int v = (i < n) ? cnt[i] : 0;
        lds[threadIdx.x] = v;
        __syncthreads();
        for (int off = 1; off < 1024; off <<= 1) {
            const int t = (threadIdx.x >= (unsigned)off) ? lds[threadIdx.x - off] : 0;
            __syncthreads();
            lds[threadIdx.x] += t;
            __syncthreads();
        }
        const int incl = lds[threadIdx.x];
        const int excl = incl - v;
        if (i < n) { rowptr[i] = carry + excl; cursor[i] = carry + excl; }
        if (threadIdx.x == 1023) carryS = carry + incl;   // chunk total
        __syncthreads();
        carry = carryS;
    }
    if (threadIdx.x == 0) rowptr[n] = carry;
}

__global__ void fill_csr(const int* __restrict__ srcA, const int* __restrict__ dstA,
                         int* __restrict__ cursor, int* __restrict__ col, int E, int n) {
    const int e = blockIdx.x * blockDim.x + threadIdx.x;
    if (e >= E + n) return;
    const int s = (e < E) ? srcA[e] : (e - E);
    const int d = (e < E) ? dstA[e] : (e - E);
    const int slot = atomicAdd(&cursor[d], 1);
    col[slot] = s;                                  // store source node per incoming edge
}

// ======== layer-1: wave per (dst,head); online softmax gather; fused bias+ReLU ======
__global__ void gat_gather_l1(const int* __restrict__ rowptr, const int* __restrict__ col,
                              const float* __restrict__ aS, const float* __restrict__ aD,
                              const float* __restrict__ H1, const float* __restrict__ b1,
                              float* __restrict__ out1, int n) {
    const int lane = threadIdx.x & 31;
    const int w = (blockIdx.x * blockDim.x + threadIdx.x) >> 5;
    if (w >= n * NHD) return;
    const int node = w >> 3, h = w & 7;
    const int beg = rowptr[node], end = rowptr[node + 1];
    const float aDd = aD[node * NHD + h];
    float m = -__builtin_inff(), den = 0.f, acc = 0.f;
    for (int j = beg; j < end; ++j) {
        const int s = col[j];
        if (j + 1 < end) {   // prefetch next source row (global_prefetch path)
            const int s2 = col[j + 1];
            __builtin_prefetch(&H1[(size_t)s2 * HC + h * CPH + lane], 0, 1);
        }
        const float l = leaky(aS[s * NHD + h] + aDd);
        const float mn = fmaxf(m, l);
        const float scale = expf(m - mn);           // first iter: exp(-inf) = 0
        const float wgt   = expf(l - mn);
        acc = acc * scale + wgt * H1[(size_t)s * HC + h * CPH + lane];
        den = den * scale + wgt;
        m = mn;
    }
    const float v = acc / den + b1[h * CPH + lane];
    out1[(size_t)node * HC + h * CPH + lane] = v > 0.f ? v : 0.f;
}

// === layer-2: wave per dst node; online softmax gather + bias + log_softmax fused ===
__global__ void gat_l2_fused(const int* __restrict__ rowptr, const int* __restrict__ col,
                             const float* __restrict__ aS, const float* __restrict__ aD,
                             const float* __restrict__ H2, const float* __restrict__ b2,
                             float* __restrict__ out, int n) {
    const int lane = threadIdx.x & 31;
    const int node = (blockIdx.x * blockDim.x + threadIdx.x) >> 5;
    if (node >= n) return;
    const int beg = rowptr[node], end = rowptr[node + 1];
    const float aDd = aD[node];
    const int c1 = lane + 32;
    float m = -__builtin_inff(), den = 0.f, acc0 = 0.f, acc1 = 0.f;
    for (int j = beg; j < end; ++j) {
        const int s = col[j];
        if (j + 1 < end) {
            const int s2 = col[j + 1];
            __builtin_prefetch(&H2[(size_t)s2 * OUTC + lane], 0, 1);
        }
        const float l = leaky(aS[s] + aDd);
        const float mn = fmaxf(m, l);
        const float scale = expf(m - mn);
        const float wgt   = expf(l - mn);
        const float* hrow = H2 + (size_t)s * OUTC;
        acc0 = acc0 * scale + wgt * ((lane < OUTC) ? hrow[lane] : 0.f);
        acc1 = acc1 * scale + wgt * ((c1 < OUTC) ? hrow[c1] : 0.f);
        den = den * scale + wgt;
        m = mn;
    }
    float v0 = (lane < OUTC) ? acc0 / den + b2[lane] : -__builtin_inff();
    float v1 = (c1 < OUTC)  ? acc1 / den + b2[c1]  : -__builtin_inff();
    // wave log_softmax over 40 classes
    float mx = fmaxf(v0, v1);
    for (int o = 16; o > 0; o >>= 1) mx = fmaxf(mx, __shfl_xor(mx, o, 32));
    float ssum = 0.f;
    if (lane < OUTC) ssum += expf(v0 - mx);
    if (c1 < OUTC)  ssum += expf(v1 - mx);
    for (int o = 16; o > 0; o >>= 1) ssum += __shfl_xor(ssum, o, 32);
    const float lse = mx + logf(ssum);
    if (lane < OUTC) out[(size_t)node * OUTC + lane] = v0 - lse;
    if (c1 < OUTC)  out[(size_t)node * OUTC + c1]  = v1 - lse;
}

extern "C" void kernel_launch(void* const* d_in, const int* in_sizes, int n_in,
                              void* d_out, int out_size, void* d_ws, size_t ws_size,
                              hipStream_t stream) {
    const float* x   = (const float*)d_in[0];
    const int*   ei  = (const int*)d_in[1];
    const float* W1  = (const float*)d_in[2];
    const float* as1 = (const float*)d_in[3];
    const float* ad1 = (const float*)d_in[4];
    const float* b1  = (const float*)d_in[5];
    const float* W2  = (const float*)d_in[6];
    const float* as2 = (const float*)d_in[7];
    const float* ad2 = (const float*)d_in[8];
    const float* b2  = (const float*)d_in[9];
    const int N  = in_sizes[0] / INCH;
    const int E  = in_sizes[1] / 2;
    const int Et = E + N;
    const int* srcA = ei;
    const int* dstA = ei + E;

    // workspace layout (4-byte units); layer-2 buffers reuse layer-1 regions
    float* ws = (float*)d_ws;
    size_t off = 0;
    float* h1   = ws + off; off += (size_t)N * HC;
    float* out1 = ws + off; off += (size_t)N * HC;
    float* aS1  = ws + off; off += (size_t)N * NHD;
    float* aD1  = ws + off; off += (size_t)N * NHD;
    int* rowptr = (int*)(ws + off); off += (size_t)N + 1;
    int* cnt    = (int*)(ws + off); off += (size_t)N;
    int* cursor = (int*)(ws + off); off += (size_t)N;
    int* col    = (int*)(ws + off); off += (size_t)Et;
    float* h2  = h1;    // N*OUTC <= N*HC; h1 dead after gat_gather_l1
    float* aS2 = aS1;
    float* aD2 = aD1;

    // ---- CSR by destination (shared by both layers) ----
    hipMemsetAsync(cnt, 0, (size_t)N * sizeof(int), stream);
    count_edges<<<(Et + 255) / 256, 256, 0, stream>>>(dstA, cnt, E, N);
    scan_rowptr<<<1, 1024, 0, stream>>>(cnt, rowptr, cursor, N);
    fill_csr<<<(Et + 255) / 256, 256, 0, stream>>>(srcA, dstA, cursor, col, E, N);

    // ---- layer 1 ----
    gemm_x_w1<<<(N + 15) / 16, 256, 0, stream>>>(x, W1, h1, N);
    attn_dots_l1<<<(N * NHD + 255) / 256, 256, 0, stream>>>(h1, as1, ad1, aS1, aD1, N);
    gat_gather_l1<<<(N * NHD * 32 + 255) / 256, 256, 0, stream>>>(
        rowptr, col, aS1, aD1, h1, b1, out1, N);

    // ---- layer 2 ----
    gemm_h_w2<<<(N + 15) / 16, 96, 0, stream>>>(out1, W2, h2, N);
    attn_dots_l2<<<(N + 255) / 256, 256, 0, stream>>>(h2, as2, ad2, aS2, aD2, N);
    gat_l2_fused<<<(N * 32 + 255) / 256, 256, 0, stream>>>(
        rowptr, col, aS2, aD2, h2, b2, (float*)d_out, N);
}